// MinPlus_71906342469870
// MI455X (gfx1250) — compile-verified
//
#include <hip/hip_runtime.h>
#include <cstdint>
#include <cstddef>

// ---------------------------------------------------------------------------
// Min-plus semiring matmul + bias for MI455X (gfx1250).
//   y[n,o] = min_i ( x[n,i] + w[o,i] ) + bias[o]
// VALU-bound problem (WMMA cannot express min-of-sums). CDNA5 features used:
//   * TENSOR_LOAD_TO_LDS (TDM) stages the 64KB weight into LDS once per block,
//     with hardware row padding (1 DWORD per 128 DWORDs) -> LDS stride 129
//     -> bank-conflict-free per-lane weight-row reads.
//   * s_wait_tensorcnt for TDM completion.
//   * v_pk_add_f32 (VOP3P packed fp32) for the adds; v_min3_num_f32 min tree.
//   * wave32 mapping: lane = output column; rows register-blocked (R=8);
//     x-row reads are wave-uniform -> SMEM/scalar path.
// ---------------------------------------------------------------------------

typedef __attribute__((ext_vector_type(4))) unsigned int u32x4;
typedef __attribute__((ext_vector_type(8))) int          i32x8;
typedef __attribute__((ext_vector_type(4))) int          i32x4;
typedef __attribute__((ext_vector_type(2))) float        v2f;

#define DIN            128
#define DOUT           128
#define LDS_STRIDE     129   // +1 DWORD pad per weight row -> banks (o+i)%64
#define R_ROWS         8     // rows per thread (register blocked)
#define ROWS_PER_BLOCK 16    // 2 row-groups * R_ROWS

__global__ __launch_bounds__(256) void MinPlus_71906342469870_kernel(
    const float* __restrict__ x,     // (nrows, 128)
    const float* __restrict__ w,     // (128, 128) row-major: w[o*128+i]
    const float* __restrict__ bias,  // (128,)
    float* __restrict__ out,         // (nrows, 128)
    int nrows)
{
    __shared__ float sw[DOUT * LDS_STRIDE];   // 66048 B, padded weight tile

    const unsigned tid  = threadIdx.x;
    const unsigned wid  = (unsigned)__builtin_amdgcn_readfirstlane((int)(tid >> 5));
    const unsigned lane = tid & 31u;

    // ---- Stage weight into LDS via the Tensor Data Mover (wave 0 only) ----
    if (wid == 0) {
        const uint32_t lds_base = (uint32_t)(uintptr_t)(&sw[0]);  // LDS byte offset
        const uint64_t ga       = (uint64_t)(uintptr_t)w;

        // D# group 0: count=1 (valid), lds_addr, global_addr, type=2 (image)
        u32x4 g0;
        g0[0] = 1u;
        g0[1] = lds_base;
        g0[2] = (uint32_t)(ga & 0xFFFFFFFFu);
        g0[3] = (uint32_t)((ga >> 32) & 0x01FFFFFFu) | (2u << 30);

        // D# group 1:
        //  [17:16] data_size=2 (4B)  [20] pad_enable=1
        //  [24:22] pad_interval=6 (every 128 DWORDs)  [31:25] pad_amount=0 (1 DWORD)
        //  tensor_dim0=128, tensor_dim1=128, tile_dim0=128, tile_dim1=128,
        //  tensor_dim0_stride=128
        i32x8 g1;
        g1[0] = (int)((2u << 16) | (1u << 20) | (6u << 22));
        g1[1] = (int)(128u << 16);   // tensor_dim0[15:0] in bits [63:48]
        g1[2] = (int)(128u << 16);   // tensor_dim1[15:0] in bits [95:80]
        g1[3] = (int)(128u << 16);   // tile_dim0 in bits [127:112]
        g1[4] = (int)(128u);         // tile_dim1 in bits [143:128]
        g1[5] = (int)(128u);         // tensor_dim0_stride low bits [191:160]
        g1[6] = 0;
        g1[7] = 0;                   // tensor_dim1_stride unused (2D tile)

        i32x4 g2 = (i32x4)0;         // 2D tensor: groups 2/3 NULL/zero
        i32x4 g3 = (i32x4)0;

#if defined(__clang_major__) && (__clang_major__ >= 23)
        i32x8 g4 = (i32x8)0;
        __builtin_amdgcn_tensor_load_to_lds(g0, g1, g2, g3, g4, 0);
#else
        __builtin_amdgcn_tensor_load_to_lds(g0, g1, g2, g3, 0);
#endif
        __builtin_amdgcn_s_wait_tensorcnt(0);
    }
    __syncthreads();

    // ---- Thread mapping: lane -> output column, register block over rows ----
    const unsigned colgrp = wid & 3u;          // 4 column groups of 32
    const unsigned rowgrp = wid >> 2;          // 2 row groups
    const unsigned o      = colgrp * 32u + lane;

    const float bo = bias[o];
    const float* __restrict__ swp = &sw[(size_t)o * LDS_STRIDE];

    const int rbase = (int)(blockIdx.x * ROWS_PER_BLOCK + rowgrp * R_ROWS);
    const float* __restrict__ xp = x + (size_t)rbase * DIN;  // wave-uniform base

    float acc[R_ROWS];
#pragma unroll
    for (int r = 0; r < R_ROWS; ++r) acc[r] = __FLT_MAX__;

    // Inner reduction, 8 k-steps per body:
    //   weight: 8 conflict-free ds_load_b32 (padded stride 129)
    //   x:      wave-uniform v2f loads (SMEM path), 8-byte aligned
    //   math:   4x v_pk_add_f32 + min tree (v_min_num/v_min3_num) per row
    for (int i = 0; i < DIN; i += 8) {
        v2f wv[4];
#pragma unroll
        for (int j = 0; j < 4; ++j) {
            wv[j].x = swp[i + 2 * j + 0];
            wv[j].y = swp[i + 2 * j + 1];
        }
#pragma unroll
        for (int r = 0; r < R_ROWS; ++r) {
            const v2f* __restrict__ xr =
                (const v2f*)(xp + (size_t)r * DIN + i);
            const v2f t0 = xr[0] + wv[0];   // v_pk_add_f32
            const v2f t1 = xr[1] + wv[1];
            const v2f t2 = xr[2] + wv[2];
            const v2f t3 = xr[3] + wv[3];
            const float m0 = fminf(t0.x, t0.y);
            const float m1 = fminf(t1.x, t1.y);
            const float m2 = fminf(t2.x, t2.y);
            const float m3 = fminf(t3.x, t3.y);
            acc[r] = fminf(acc[r], fminf(m0, m1));  // -> v_min3_num_f32
            acc[r] = fminf(acc[r], fminf(m2, m3));  // -> v_min3_num_f32
        }
    }

    // ---- Epilogue: bias add + coalesced store (lane -> consecutive o) ----
#pragma unroll
    for (int r = 0; r < R_ROWS; ++r) {
        const int n = rbase + r;
        if (n < nrows) out[(size_t)n * DIN + o] = acc[r] + bo;
    }
}

extern "C" void kernel_launch(void* const* d_in, const int* in_sizes, int n_in,
                              void* d_out, int out_size, void* d_ws, size_t ws_size,
                              hipStream_t stream) {
    (void)n_in; (void)out_size; (void)d_ws; (void)ws_size;
    const float* x    = (const float*)d_in[0];
    const float* w    = (const float*)d_in[1];
    const float* bias = (const float*)d_in[2];
    float*       out  = (float*)d_out;

    const int nrows = in_sizes[0] / DIN;                      // 16*2048 = 32768
    const int grid  = (nrows + ROWS_PER_BLOCK - 1) / ROWS_PER_BLOCK;  // 2048

    MinPlus_71906342469870_kernel<<<grid, 256, 0, stream>>>(x, w, bias, out, nrows);
}